// VectorQuantizer_7447473291875
// MI455X (gfx1250) — compile-verified
//
#include <hip/hip_runtime.h>
#include <hip/hip_bf16.h>

// ---------------------------------------------------------------------------
// VQ-VAE vector quantizer for MI455X (gfx1250, wave32, WMMA).
//   N = 65536 tokens, D = 128, K = 1024 codes.
//   Dominant cost: Z @ E^T (17.2 GFLOP fp32) -> V_WMMA_F32_16X16X4_F32,
//   with codebook tiles double-buffered in LDS via GLOBAL_LOAD_ASYNC_TO_LDS
//   (ASYNCcnt) — builtin present on this toolchain (signature: v4i AS1*,
//   v4i AS3*, imm offset, imm cpol, per the round-2 diagnostic).
// ---------------------------------------------------------------------------

typedef __attribute__((ext_vector_type(2))) float v2f;
typedef __attribute__((ext_vector_type(8))) float v8f;

// Async-to-LDS builtin pointer types: 4 x i32 GCC vector, AS1 (global) src,
// AS3 (LDS) dst.
typedef int v4i __attribute__((vector_size(16)));
typedef __attribute__((address_space(1))) v4i* as1_v4i;
typedef __attribute__((address_space(3))) v4i* as3_v4i;

#define VQ_D 128
#define VQ_K 1024
#define DECAY_F 0.99f
#define ONE_MINUS_DECAY_F 0.01f
#define EPS_F 1e-5f

// LDS tile: 16 codebook rows x 128 dims, padded to stride 132 floats
// (528 B = 33*16 B -> 16B-aligned rows; bank = (n*4 + k) % 64 -> conflict-free
//  b64 B-fragment reads across each half-wave).
#define E_STRIDE 132

#if defined(__has_builtin)
#if __has_builtin(__builtin_amdgcn_global_load_async_to_lds_b128) && \
    __has_builtin(__builtin_amdgcn_s_wait_asynccnt)
#define VQ_HAVE_ASYNC 1
#endif
#endif

// ------------------------- kernel 0: zero scratch ---------------------------
__global__ void vq_zero_kernel(float* p, int n) {
    int g = blockIdx.x * blockDim.x + threadIdx.x;
    if (g < n) p[g] = 0.0f;
}

// ------------------------- kernel 1: ||e_k||^2 ------------------------------
__global__ void vq_enorm_kernel(const float* __restrict__ emb,
                                float* __restrict__ enorm) {
    int k = blockIdx.x * blockDim.x + threadIdx.x;
    if (k >= VQ_K) return;
    const float* e = emb + (size_t)k * VQ_D;
    float s = 0.0f;
#pragma unroll 8
    for (int d = 0; d < VQ_D; ++d) { float v = e[d]; s += v * v; }
    enorm[k] = s;
}

// ---------------- kernel 2: WMMA distance + argmin --------------------------
// Block = 256 threads = 8 waves. Each wave owns a 16-row tile of z.
// Score(m,n) = ||e_n||^2 - 2 * (z_m . e_n); ||z_m||^2 dropped (const per row).
__global__ __launch_bounds__(256) void vq_argmin_kernel(
    const float* __restrict__ z, const float* __restrict__ emb,
    const float* __restrict__ enorm_g, int* __restrict__ idx_out,
    float* __restrict__ idx_f_out) {
    __shared__ float lds_e[2][16 * E_STRIDE];
    __shared__ float lds_en[2][16];

    const int wave = threadIdx.x >> 5;      // 0..7
    const int lane = threadIdx.x & 31;
    const int rowbase = blockIdx.x * 128 + wave * 16;
    const int m = lane & 15;                // row within 16-row tile
    const int khalf = (lane >> 4) * 2;      // lanes 0-15: K={0,1}; 16-31: K={2,3}

    // A fragments (16x4 fp32 per WMMA), all 32 K-chunks resident in VGPRs.
    v2f a[32];
    {
        const float* zrow = z + (size_t)(rowbase + m) * VQ_D;
#pragma unroll
        for (int c = 0; c < 32; ++c) {
            const int kk = c * 4 + khalf;
            a[c].x = zrow[kk];
            a[c].y = zrow[kk + 1];
        }
    }

    float bestv[8];
    int   besti[8];
#pragma unroll
    for (int r = 0; r < 8; ++r) { bestv[r] = 3.4e38f; besti[r] = 0; }

    const int n = lane & 15;                // code column within tile

    // Stage codebook tile t into LDS buffer b.
    auto stage = [&](int t, int b) {
        const int cb = t * 16;
        const float* src = emb + (size_t)cb * VQ_D;
#ifdef VQ_HAVE_ASYNC
        // 512 x 16B chunks, 2 async b128 per thread: global -> LDS directly,
        // no VGPR round-trip; tracked with ASYNCcnt.
#pragma unroll
        for (int j = (int)threadIdx.x; j < 512; j += 256) {
            __builtin_amdgcn_global_load_async_to_lds_b128(
                (as1_v4i)(src + j * 4),
                (as3_v4i)&lds_e[b][(j >> 5) * E_STRIDE + (j & 31) * 4],
                0, 0);
        }
#else
#pragma unroll
        for (int j = (int)threadIdx.x; j < 512; j += 256) {
            float4 v = ((const float4*)src)[j];
            *(float4*)&lds_e[b][(j >> 5) * E_STRIDE + (j & 31) * 4] = v;
        }
#endif
        if (threadIdx.x < 16) lds_en[b][threadIdx.x] = enorm_g[cb + threadIdx.x];
    };
    // Own async slice done -> then block-wide fence (syncthreads also waits
    // dscnt, covering the non-async staging path and the enorm ds_store).
    auto wait_stage = [&]() {
#ifdef VQ_HAVE_ASYNC
        __builtin_amdgcn_s_wait_asynccnt(0);
#endif
        __syncthreads();
    };

    stage(0, 0);
    wait_stage();

    for (int t = 0; t < VQ_K / 16; ++t) {
        const int cur = t & 1;
        if (t + 1 < VQ_K / 16) stage(t + 1, cur ^ 1);      // overlap with WMMA
        if (t + 2 < VQ_K / 16)                             // global_prefetch_b8
            __builtin_prefetch(emb + (size_t)(t + 2) * 16 * VQ_D, 0, 1);

        // 32 chained fp32 WMMAs: acc(16x16) += A(16x4) x B(4x16)
        v8f acc = {};
        const float* e = lds_e[cur];
#pragma unroll
        for (int c = 0; c < 32; ++c) {
            const int kk = c * 4 + khalf;
            v2f b;
            b.x = e[n * E_STRIDE + kk];
            b.y = e[n * E_STRIDE + kk + 1];
            acc = __builtin_amdgcn_wmma_f32_16x16x4_f32(
                false, a[c], false, b, (short)0, acc, false, false);
        }

        const float en = lds_en[cur][n];
        const int cb = t * 16;
#pragma unroll
        for (int r = 0; r < 8; ++r) {
            const float s = en - 2.0f * acc[r];
            const int ci = cb + n;
            if (s < bestv[r] || (s == bestv[r] && ci < besti[r])) {
                bestv[r] = s; besti[r] = ci;
            }
        }

        if (t + 1 < VQ_K / 16) wait_stage();
    }

    // C layout: VGPR r holds M=r (lanes 0-15) and M=r+8 (lanes 16-31).
    // Reduce the 16 code-candidates within each half-wave; first-index tiebreak.
#pragma unroll
    for (int r = 0; r < 8; ++r) {
        float v = bestv[r]; int i = besti[r];
#pragma unroll
        for (int mk = 8; mk >= 1; mk >>= 1) {
            const float ov = __shfl_xor(v, mk, 32);
            const int   oi = __shfl_xor(i, mk, 32);
            if (ov < v || (ov == v && oi < i)) { v = ov; i = oi; }
        }
        if ((lane & 15) == 0) {
            const int row = rowbase + r + ((lane >> 4) ? 8 : 0);
            idx_out[row] = i;
            idx_f_out[row] = (float)i;
        }
    }
}

// ---------- kernel 3: gather quantized + loss partial + segment sums --------
// One thread per 4 dims (float4). N*D/4 threads.
__global__ __launch_bounds__(256) void vq_gather_kernel(
    const float* __restrict__ z, const float* __restrict__ emb,
    const int* __restrict__ idx, float* __restrict__ q_out,
    float* __restrict__ counts, float* __restrict__ dw,
    float* __restrict__ loss_sum) {
    const int tid = blockIdx.x * 256 + threadIdx.x;
    const int row = tid >> 5;          // /(D/4)
    const int c4  = tid & 31;
    const int k = idx[row];

    const float4 z4 = ((const float4*)z)[tid];
    const float4 e4 = ((const float4*)(emb + (size_t)k * VQ_D))[c4];
    ((float4*)q_out)[tid] = e4;        // quantized_st == quantized numerically

    const float dx = z4.x - e4.x, dy = z4.y - e4.y;
    const float dz = z4.z - e4.z, dv = z4.w - e4.w;
    float s = dx * dx + dy * dy + dz * dz + dv * dv;

    float* dwp = dw + (size_t)k * VQ_D + c4 * 4;
    atomicAdd(dwp + 0, z4.x);
    atomicAdd(dwp + 1, z4.y);
    atomicAdd(dwp + 2, z4.z);
    atomicAdd(dwp + 3, z4.w);
    if (c4 == 0) atomicAdd(&counts[k], 1.0f);

    __shared__ float red[256];
    red[threadIdx.x] = s;
    __syncthreads();
    for (int off = 128; off > 0; off >>= 1) {
        if (threadIdx.x < off) red[threadIdx.x] += red[threadIdx.x + off];
        __syncthreads();
    }
    if (threadIdx.x == 0) atomicAdd(loss_sum, red[0]);
}

// ------- kernel 4: new_cluster_size, n = sum, loss finalize (1 block) -------
__global__ __launch_bounds__(256) void vq_ema_a_kernel(
    const float* __restrict__ ema_cs, const float* __restrict__ counts,
    const float* __restrict__ loss_sum, float* __restrict__ ncs_out,
    float* __restrict__ n_out, float* __restrict__ loss_out, float inv_ND) {
    float local = 0.0f;
    for (int k = threadIdx.x; k < VQ_K; k += 256) {
        const float v = DECAY_F * ema_cs[k] + ONE_MINUS_DECAY_F * counts[k];
        ncs_out[k] = v;
        local += v;
    }
    __shared__ float red[256];
    red[threadIdx.x] = local;
    __syncthreads();
    for (int off = 128; off > 0; off >>= 1) {
        if (threadIdx.x < off) red[threadIdx.x] += red[threadIdx.x + off];
        __syncthreads();
    }
    if (threadIdx.x == 0) {
        n_out[0] = red[0];
        // loss = COMMITMENT*mean + mean = 1.25 * sum / (N*D)
        loss_out[0] = 1.25f * loss_sum[0] * inv_ND;
    }
}

// ------- kernel 5: new_ema_w and normalized new_emb_weight (K*D) ------------
__global__ __launch_bounds__(256) void vq_ema_b_kernel(
    const float* __restrict__ ema_w, const float* __restrict__ dw,
    const float* __restrict__ ncs, const float* __restrict__ n_val,
    float* __restrict__ emaw_out, float* __restrict__ emb_out) {
    const int g = blockIdx.x * 256 + threadIdx.x;
    if (g >= VQ_K * VQ_D) return;
    const int k = g >> 7;  // /D
    const float nw = DECAY_F * ema_w[g] + ONE_MINUS_DECAY_F * dw[g];
    emaw_out[g] = nw;
    const float nn = n_val[0];
    const float cs = (ncs[k] + EPS_F) / (nn + (float)VQ_K * EPS_F) * nn;
    emb_out[g] = nw / cs;
}

// ---------------------------------------------------------------------------
extern "C" void kernel_launch(void* const* d_in, const int* in_sizes, int n_in,
                              void* d_out, int out_size, void* d_ws, size_t ws_size,
                              hipStream_t stream) {
    const float* z       = (const float*)d_in[0];   // [N, 128]
    const float* emb_w   = (const float*)d_in[1];   // [1024, 128]
    const float* ema_cs  = (const float*)d_in[2];   // [1024]
    const float* ema_w   = (const float*)d_in[3];   // [1024, 128]

    const int N  = in_sizes[0] / VQ_D;              // 65536
    const int ND = N * VQ_D;

    // d_out layout (tuple, flat, return order):
    float* out = (float*)d_out;
    float* q_out    = out;                          // [N, D]
    float* loss_out = out + (size_t)ND;             // [1]
    float* idxf_out = loss_out + 1;                 // [N]
    float* emb_out  = idxf_out + N;                 // [K, D]
    float* ncs_out  = emb_out + (size_t)VQ_K * VQ_D;// [K]
    float* emaw_out = ncs_out + VQ_K;               // [K, D]

    // d_ws layout:
    int*   idx_ws   = (int*)d_ws;                   // [N] int32
    float* wsf      = (float*)d_ws;
    float* enorm_ws = wsf + N;                      // [K]
    float* counts_ws= enorm_ws + VQ_K;              // [K]
    float* dw_ws    = counts_ws + VQ_K;             // [K, D]
    float* loss_ws  = dw_ws + (size_t)VQ_K * VQ_D;  // [1]
    float* n_ws     = loss_ws + 1;                  // [1]

    // 0: zero counts + dw + loss accumulator
    {
        const int nz = VQ_K + VQ_K * VQ_D + 1;
        vq_zero_kernel<<<(nz + 255) / 256, 256, 0, stream>>>(counts_ws, nz);
    }
    // 1: codebook norms
    vq_enorm_kernel<<<(VQ_K + 255) / 256, 256, 0, stream>>>(emb_w, enorm_ws);
    // 2: WMMA distances + argmin (128 rows per block)
    vq_argmin_kernel<<<N / 128, 256, 0, stream>>>(z, emb_w, enorm_ws,
                                                  idx_ws, idxf_out);
    // 3: gather + loss partials + segment sums
    vq_gather_kernel<<<(ND / 4) / 256, 256, 0, stream>>>(
        z, emb_w, idx_ws, q_out, counts_ws, dw_ws, loss_ws);
    // 4: cluster-size EMA + n + loss finalize
    vq_ema_a_kernel<<<1, 256, 0, stream>>>(ema_cs, counts_ws, loss_ws,
                                           ncs_out, n_ws, loss_out,
                                           1.0f / (float)ND);
    // 5: ema_w EMA + normalized codebook
    vq_ema_b_kernel<<<(VQ_K * VQ_D + 255) / 256, 256, 0, stream>>>(
        ema_w, dw_ws, ncs_out, n_ws, emaw_out, emb_out);
}